// RateFit_72541997629951
// MI455X (gfx1250) — compile-verified
//
#include <hip/hip_runtime.h>
#include <hip/hip_bf16.h>
#include <stdint.h>

#define BETA_C 0.1f
#define RES_C  100

typedef __attribute__((ext_vector_type(2))) float    v2f;
typedef __attribute__((ext_vector_type(8))) float    v8f;
typedef __attribute__((ext_vector_type(4))) unsigned v4u;
typedef __attribute__((ext_vector_type(8))) int      v8i;
typedef __attribute__((ext_vector_type(4))) int      v4i;

// ---------------------------------------------------------------------------
// K0: scalar reductions: tmin, tmax, max(diff(t_init)), sum(init_conc)
// ---------------------------------------------------------------------------
__global__ void __launch_bounds__(1024)
scalars_kernel(const float* __restrict__ t_init, const float* __restrict__ init_conc,
               int TN, int C, float* __restrict__ scal) {
  __shared__ float red[4][32];
  int tid = threadIdx.x;
  float mn = 3.4e38f, mx = -3.4e38f, md = -3.4e38f, sm = 0.f;
  for (int i = tid; i < TN; i += blockDim.x) {
    float v = t_init[i];
    mn = fminf(mn, v);
    mx = fmaxf(mx, v);
    if (i + 1 < TN) md = fmaxf(md, t_init[i + 1] - v);
  }
  for (int i = tid; i < C; i += blockDim.x) sm += init_conc[i];
  for (int o = 16; o; o >>= 1) {
    mn = fminf(mn, __shfl_down(mn, o, 32));
    mx = fmaxf(mx, __shfl_down(mx, o, 32));
    md = fmaxf(md, __shfl_down(md, o, 32));
    sm += __shfl_down(sm, o, 32);
  }
  int l = tid & 31, w = tid >> 5;
  if (l == 0) { red[0][w] = mn; red[1][w] = mx; red[2][w] = md; red[3][w] = sm; }
  __syncthreads();
  if (tid == 0) {
    int nw = blockDim.x >> 5;
    float a = red[0][0], b = red[1][0], c = red[2][0], d = red[3][0];
    for (int i = 1; i < nw; ++i) {
      a = fminf(a, red[0][i]); b = fmaxf(b, red[1][i]);
      c = fmaxf(c, red[2][i]); d += red[3][i];
    }
    scal[0] = a; scal[1] = b; scal[2] = c; scal[3] = d;
  }
}

// ---------------------------------------------------------------------------
// K1: compact nu (4096x2048, 4 nnz/row) -> cols[R][4], vals[R][4]
// One wave per row; coalesced 128B/wave loads; single pass over 32 MB.
// ---------------------------------------------------------------------------
__global__ void __launch_bounds__(256)
extract_kernel(const float* __restrict__ nu, int R, int C,
               int* __restrict__ cols, float* __restrict__ vals) {
  __shared__ int   cnt[8];
  __shared__ int   scol[8][4];
  __shared__ float sval[8][4];
  int w = threadIdx.x >> 5, l = threadIdx.x & 31;
  int row = blockIdx.x * 8 + w;
  if (l == 0) cnt[w] = 0;
  __syncthreads();
  if (row < R) {
    const float* nr = nu + (size_t)row * C;
    for (int c = l; c < C; c += 32) {
      __builtin_prefetch(nr + c + 512, 0, 0);   // global_prefetch_b8
      float v = nr[c];
      if (v != 0.f) {
        int s = atomicAdd(&cnt[w], 1);
        if (s < 4) { scol[w][s] = c; sval[w][s] = v; }
      }
    }
  }
  __syncthreads();
  if (threadIdx.x < 8) {
    int rr = blockIdx.x * 8 + threadIdx.x;
    if (rr < R) {
      int n = min(cnt[threadIdx.x], 4);
      for (int s = 0; s < 4; ++s) {
        cols[rr * 4 + s] = (s < n) ? scol[threadIdx.x][s] : 0;
        vals[rr * 4 + s] = (s < n) ? sval[threadIdx.x][s] : 0.f;
      }
    }
  }
}

// ---------------------------------------------------------------------------
// K2: sequential explicit-Euler integration, single workgroup.
// State + rate accumulator live in LDS; per-reaction constants hoisted to
// registers before the 99-step loop so each step is pure LDS traffic.
// ---------------------------------------------------------------------------
__global__ void __launch_bounds__(1024)
integrate_kernel(const float* __restrict__ f_in, const float* __restrict__ r_in,
                 const float* __restrict__ init_conc,
                 const int* __restrict__ cols, const float* __restrict__ vals,
                 const float* __restrict__ scal,
                 int R, int C, int TN, float* __restrict__ curve) {
  __shared__ float conc[2048];
  __shared__ float dC[2048];
  int tid = threadIdx.x;
  float tmin = scal[0], tmax = scal[1], maxdiff = scal[2], sumC = scal[3];
  float Kub = (float)TN * 10.0f / maxdiff;
  float dtg = (tmax - tmin) / (float)(RES_C - 1);

  const int RPT = 8;
  int4   rcols[RPT];
  float4 rvals[RPT];
  float  rkp[RPT], rkm[RPT];
  int nmine = 0;
  for (int r = tid; r < R && nmine < RPT; r += blockDim.x) {
    int4   cc = ((const int4*)cols)[r];
    float4 vv = ((const float4*)vals)[r];
    float of  = fmaxf(vv.x, 0.f) + fmaxf(vv.y, 0.f) + fmaxf(vv.z, 0.f) + fmaxf(vv.w, 0.f);
    float orr = -(fminf(vv.x, 0.f) + fminf(vv.y, 0.f) + fminf(vv.z, 0.f) + fminf(vv.w, 0.f));
    float ub0 = Kub / __powf(sumC, of - 1.f);
    float ub1 = Kub / __powf(sumC, orr - 1.f);
    // smooth_clamp(x, 0, ub) = ub * sigmoid(BETA*(x - ub/2))
    rkp[nmine] = ub0 / (1.f + __expf(-BETA_C * (f_in[r] - 0.5f * ub0)));
    rkm[nmine] = ub1 / (1.f + __expf(-BETA_C * (r_in[r] - 0.5f * ub1)));
    rcols[nmine] = cc; rvals[nmine] = vv; ++nmine;
  }

  for (int i = tid; i < C; i += blockDim.x) {
    float c0 = init_conc[i];
    conc[i] = c0; dC[i] = 0.f;
    curve[i] = c0;                       // row 0 of trajectory
  }
  __syncthreads();

  for (int s = 1; s < RES_C; ++s) {
    for (int k = 0; k < nmine; ++k) {
      float fwd = rkp[k], rev = rkm[k];
      int4 cc = rcols[k]; float4 vv = rvals[k];
      auto term = [&](int ci, float vi) {
        float c = fminf(fmaxf(conc[ci], 1e-8f), 1000.f);
        if (vi < 0.f)      fwd *= (vi == -1.f) ? c : __powf(c, -vi);
        else if (vi > 0.f) rev *= (vi ==  1.f) ? c : __powf(c,  vi);
      };
      term(cc.x, vv.x); term(cc.y, vv.y); term(cc.z, vv.z); term(cc.w, vv.w);
      float rate = fwd - rev;
      if (vv.x != 0.f) atomicAdd(&dC[cc.x], vv.x * rate);   // ds_add_f32
      if (vv.y != 0.f) atomicAdd(&dC[cc.y], vv.y * rate);
      if (vv.z != 0.f) atomicAdd(&dC[cc.z], vv.z * rate);
      if (vv.w != 0.f) atomicAdd(&dC[cc.w], vv.w * rate);
    }
    __syncthreads();
    for (int i = tid; i < C; i += blockDim.x) {
      float cn = conc[i] + dC[i] * dtg;
      conc[i] = cn; dC[i] = 0.f;
      curve[(size_t)s * C + i] = cn;
    }
    __syncthreads();
  }
}

// ---------------------------------------------------------------------------
// K3: interpolation as WMMA f32 16x16x4 tiles.
// 16 sorted queries share a <=3-cell stencil window -> A = 16x4 weights,
// B = 4x16 curve slab (TDM-staged to LDS), D = 16x16 output tile.
// Store-bound: 32 MB of f32 output.
// ---------------------------------------------------------------------------
__global__ void __launch_bounds__(256)
interp_kernel(const float* __restrict__ tq, const float* __restrict__ scal,
              const float* __restrict__ curve, int TQ, int C,
              float* __restrict__ out) {
  __shared__ float slab[4 * 128];
  __shared__ int   sj[16];
  __shared__ float sf[16];
  __shared__ int   sbase;
  int tid = threadIdx.x;
  int l = tid & 31, w = tid >> 5;
  float tmin = scal[0], tmax = scal[1];
  float dtg = (tmax - tmin) / (float)(RES_C - 1);
  int qBase   = blockIdx.x * 16;
  int colBase = blockIdx.y * 128;

  if (tid < 16) {
    float x = (tq[qBase + tid] - tmin) / dtg;
    int j = (int)floorf(x);
    j = max(0, min(j, RES_C - 2));
    sj[tid] = j;
    sf[tid] = x - (float)j;
  }
  __syncthreads();
  if (tid == 0) {
    int b = sj[0];
    for (int i = 1; i < 16; ++i) b = min(b, sj[i]);
    sbase = b;
  }
  __syncthreads();
  int base = sbase;

  if (w == 0) {
    // TDM: 2D tile (tile_dim0=128 cols, tile_dim1=4 rows) of curve -> LDS slab
    unsigned lds_off = (unsigned)(uintptr_t)(&slab[0]);      // low 32b = LDS offset
    unsigned long long ga =
        (unsigned long long)(uintptr_t)(curve + (size_t)base * C + colBase);
    v4u g0; v8i g1; v4i gz4; v8i gz8;
    g0[0] = 1u;                                              // count=1, user mode
    g0[1] = lds_off;                                         // lds_addr
    g0[2] = (unsigned)(ga & 0xFFFFFFFFu);                    // global_addr[31:0]
    g0[3] = (unsigned)((ga >> 32) & 0x1FFFFFFu) | (2u << 30);// addr[56:32] | type=2
    g1[0] = (int)(2u << 16);                                 // data_size=4B, mask=0
    g1[1] = (int)(((unsigned)C & 0xFFFFu) << 16);            // tensor_dim0 lo16
    g1[2] = (int)((((unsigned)C >> 16) & 0xFFFFu)
                  | (((unsigned)RES_C & 0xFFFFu) << 16));    // dim0 hi | dim1 lo
    g1[3] = (int)(128u << 16);                               // dim1 hi=0 | tile_dim0=128
    g1[4] = (int)4u;                                         // tile_dim1=4, tile_dim2=0
    g1[5] = (int)(unsigned)C;                                // tensor_dim0_stride lo32
    g1[6] = 0; g1[7] = 0;
    gz4[0] = gz4[1] = gz4[2] = gz4[3] = 0;
    gz8[0] = gz8[1] = gz8[2] = gz8[3] = 0;
    gz8[4] = gz8[5] = gz8[6] = gz8[7] = 0;
    // 6-arg variant (clang-23 / therock-10.0 headers): groups 0..3 + spare + cpol
    __builtin_amdgcn_tensor_load_to_lds(g0, g1, gz4, gz4, gz8, 0);
    __builtin_amdgcn_s_wait_tensorcnt(0);
  }
  __syncthreads();

  // A: 16x4 f32 (lanes 0-15: M=lane, K0/K1; lanes 16-31: M=lane-16, K2/K3)
  int m  = l & 15;
  int kp = (l < 16) ? 0 : 2;
  float wgt[4] = {0.f, 0.f, 0.f, 0.f};
  int rel = sj[m] - base; rel = min(rel, 2);
  float fr = sf[m];
  wgt[rel]     += 1.f - fr;
  wgt[rel + 1] += fr;
  v2f a; a.x = wgt[kp]; a.y = wgt[kp + 1];

  // B: 4x16 f32 curve slab slice for this wave's 16 columns
  int cc = w * 16 + m;
  v2f b; b.x = slab[kp * 128 + cc]; b.y = slab[(kp + 1) * 128 + cc];

  v8f acc = {};
  acc = __builtin_amdgcn_wmma_f32_16x16x4_f32(
      /*neg_a=*/false, a, /*neg_b=*/false, b,
      /*c_mod=*/(short)0, acc, /*reuse_a=*/false, /*reuse_b=*/false);

  int rowOff = (l < 16) ? 0 : 8;
  int col = colBase + cc;
  for (int v = 0; v < 8; ++v)
    out[(size_t)(qBase + v + rowOff) * C + col] = acc[v];
}

// ---------------------------------------------------------------------------
extern "C" void kernel_launch(void* const* d_in, const int* in_sizes, int n_in,
                              void* d_out, int out_size, void* d_ws, size_t ws_size,
                              hipStream_t stream) {
  const float* t  = (const float*)d_in[0];
  const float* f  = (const float*)d_in[1];
  const float* r  = (const float*)d_in[2];
  const float* nu = (const float*)d_in[3];
  const float* c0 = (const float*)d_in[4];
  const float* ti = (const float*)d_in[5];
  int TQ = in_sizes[0], R = in_sizes[1], C = in_sizes[4], TN = in_sizes[5];

  char*  ws    = (char*)d_ws;
  float* scal  = (float*)ws;                                          // 16 B used (256 B slot)
  int*   cols  = (int*)(ws + 256);                                    // R*4 ints
  float* vals  = (float*)(ws + 256 + (size_t)R * 4 * sizeof(int));    // R*4 floats
  float* curve = (float*)(ws + 256 + (size_t)R * 4 * 8);              // RES*C floats
  float* outp  = (float*)d_out;

  scalars_kernel<<<1, 1024, 0, stream>>>(ti, c0, TN, C, scal);
  extract_kernel<<<(R + 7) / 8, 256, 0, stream>>>(nu, R, C, cols, vals);
  integrate_kernel<<<1, 1024, 0, stream>>>(f, r, c0, cols, vals, scal, R, C, TN, curve);
  dim3 g(TQ / 16, C / 128);
  interp_kernel<<<g, 256, 0, stream>>>(t, scal, curve, TQ, C, outp);
}